// GraphEncoder_72507637891223
// MI455X (gfx1250) — compile-verified
//
#include <hip/hip_runtime.h>
#include <math.h>

// ---------- types for WMMA fragments (wave32, gfx1250) ----------
typedef __attribute__((ext_vector_type(2))) float v2f;
typedef __attribute__((ext_vector_type(8))) float v8f;

#define HID 128
#define NHEAD 4
#define DHEAD 32

__device__ __forceinline__ float gelu_tanh(float x) {
    float x3 = x * x * x;
    return 0.5f * x * (1.0f + tanhf(0.7978845608028654f * (x + 0.044715f * x3)));
}

// float atomic max via signed/unsigned int monotonicity trick
__device__ __forceinline__ void atomicMaxFloat(float* addr, float val) {
    if (!signbit(val)) {
        atomicMax((int*)addr, __float_as_int(val));
    } else {
        atomicMin((unsigned int*)addr, (unsigned int)__float_as_int(val));
    }
}

// ---------------------------------------------------------------------------
// C[M x 128] = act(A[M x KD] @ W[KD x 128] + bias) (+ res)
// One wave per 16-row strip; 8 column tiles of 16; V_WMMA_F32_16X16X4_F32.
// W is staged transposed in LDS once per block: Wl[col*S + k], S = KDP+4 so
// bank = (4*col + k) % 64 -> the two lane-halves (k=ka vs k=ka+2) use disjoint
// banks and each B fragment is a single conflict-free ds_load_b64.
// A frag: lane L -> row L%16, k = k0 + (L/16)*2 + j  (contiguous float2, VMEM)
// D frag: lane L, vgpr r -> row base + (L/16)*8 + r, col tile*16 + L%16
// ---------------------------------------------------------------------------
template <int KD>
__global__ __launch_bounds__(256) void gemm_wmma(
    const float* __restrict__ A, const float* __restrict__ W,
    const float* __restrict__ bias, const float* __restrict__ res,
    float* __restrict__ C, int M, int act) {
    constexpr int KDP = (KD + 3) & ~3;   // K padded to multiple of 4
    constexpr int S = KDP + 4;           // LDS stride (floats): conflict-free
    __shared__ float Wl[HID * S];

    // cooperative transposed staging of W (zero-pad rows KD..KDP)
    for (int i = threadIdx.x; i < HID * KDP / 4; i += 256) {
        const int k = i >> 5;             // W row (k index)
        const int c4 = (i & 31) * 4;      // starting col
        float4 w;
        if (k < KD) {
            w = *(const float4*)(W + (size_t)k * HID + c4);
        } else {
            w.x = w.y = w.z = w.w = 0.0f;
        }
        Wl[(c4 + 0) * S + k] = w.x;
        Wl[(c4 + 1) * S + k] = w.y;
        Wl[(c4 + 2) * S + k] = w.z;
        Wl[(c4 + 3) * S + k] = w.w;
    }
    __syncthreads();

    const int lane = threadIdx.x & 31;
    const int wave = threadIdx.x >> 5;
    const int rowBase = (blockIdx.x * 8 + wave) * 16;
    if (rowBase >= M) return;  // wave-uniform exit: EXEC stays all-ones below

    const int half = lane >> 4;  // 0 or 1
    const int lr = lane & 15;
    int rowA = rowBase + lr;
    if (rowA >= M) rowA = M - 1;  // in-bounds dup reads; stores guarded later

    v8f acc[8] = {};
    const float* __restrict__ Arow = A + (size_t)rowA * KD;

    for (int k0 = 0; k0 < KDP; k0 += 4) {
        const int ka = k0 + half * 2;
        v2f afrag;
        if constexpr ((KD & 3) == 0) {
            afrag.x = Arow[ka];
            afrag.y = Arow[ka + 1];
        } else {
            // only the final unrolled step keeps live predication
            afrag.x = (ka < KD) ? Arow[ka] : 0.0f;
            afrag.y = (ka + 1 < KD) ? Arow[ka + 1] : 0.0f;
        }
#pragma unroll
        for (int t = 0; t < 8; ++t) {
            const int col = t * 16 + lr;
            // one ds_load_b64: W[ka][col], W[ka+1][col] (zero-padded past KD)
            v2f bfrag = *(const v2f*)(&Wl[col * S + ka]);
            acc[t] = __builtin_amdgcn_wmma_f32_16x16x4_f32(
                false, afrag, false, bfrag, (short)0, acc[t], false, false);
        }
    }

#pragma unroll
    for (int t = 0; t < 8; ++t) {
        const int col = t * 16 + lr;
        const float b = bias ? bias[col] : 0.0f;
#pragma unroll
        for (int r = 0; r < 8; ++r) {
            const int row = rowBase + half * 8 + r;
            if (row < M) {
                float v = acc[t][r] + b;
                if (act) v = gelu_tanh(v);
                if (res) v += res[(size_t)row * HID + col];
                C[(size_t)row * HID + col] = v;
            }
        }
    }
}

// ---------------- attention phase ----------------

__global__ void init_md(float* __restrict__ mx, float* __restrict__ dn, int n) {
    int i = blockIdx.x * blockDim.x + threadIdx.x;
    if (i < n) {
        mx[i] = -1e30f;
        dn[i] = 0.0f;
    }
}

// alpha[e,h] = dot(q[dst], k[src]+e) / sqrt(32); atomic segment max into mx[dst,h]
__global__ void attn_alpha(const int* __restrict__ src, const int* __restrict__ dst,
                           const float* __restrict__ Q, const float* __restrict__ K,
                           const float* __restrict__ Em, float* __restrict__ alpha,
                           float* __restrict__ mx, int E) {
    int idx = blockIdx.x * blockDim.x + threadIdx.x;
    if (idx >= E * NHEAD) return;
    const int e = idx >> 2, hd = idx & 3;
    const int s = src[e], d = dst[e];
    const float4* q = (const float4*)(Q + (size_t)d * HID + hd * DHEAD);
    const float4* k = (const float4*)(K + (size_t)s * HID + hd * DHEAD);
    const float4* ev = (const float4*)(Em + (size_t)e * HID + hd * DHEAD);
    float a = 0.0f;
#pragma unroll
    for (int i = 0; i < 8; ++i) {
        float4 qv = q[i], kv = k[i], em = ev[i];
        a += qv.x * (kv.x + em.x) + qv.y * (kv.y + em.y) +
             qv.z * (kv.z + em.z) + qv.w * (kv.w + em.w);
    }
    a *= 0.17677669529663687f;  // 1/sqrt(32)
    alpha[idx] = a;
    atomicMaxFloat(&mx[d * NHEAD + hd], a);
}

// ex = exp(alpha - mx[dst]); store in place; atomic add into denom
__global__ void attn_exp(const int* __restrict__ dst, float* __restrict__ alpha,
                         const float* __restrict__ mx, float* __restrict__ dn, int E) {
    int idx = blockIdx.x * blockDim.x + threadIdx.x;
    if (idx >= E * NHEAD) return;
    const int e = idx >> 2, hd = idx & 3;
    const int d = dst[e];
    float ex = __expf(alpha[idx] - mx[d * NHEAD + hd]);
    alpha[idx] = ex;
    atomicAdd(&dn[d * NHEAD + hd], ex);
}

// h[dst] += (ex/denom) * (v[src] + e); one thread per (edge, 4-float quad)
__global__ void attn_scatter(const int* __restrict__ src, const int* __restrict__ dst,
                             const float* __restrict__ V, const float* __restrict__ Em,
                             const float* __restrict__ alpha, const float* __restrict__ dn,
                             float* __restrict__ H, int E) {
    long long idx = (long long)blockIdx.x * blockDim.x + threadIdx.x;
    if (idx >= (long long)E * 32) return;
    const int e = (int)(idx >> 5);
    const int q = (int)(idx & 31);
    const int hd = q >> 3;
    const int s = src[e], d = dst[e];
    const float w = alpha[e * NHEAD + hd] / dn[d * NHEAD + hd];
    float4 v = *(const float4*)(V + (size_t)s * HID + q * 4);
    float4 ev = *(const float4*)(Em + (size_t)e * HID + q * 4);
    float* out = H + (size_t)d * HID + q * 4;
    atomicAdd(out + 0, w * (v.x + ev.x));
    atomicAdd(out + 1, w * (v.y + ev.y));
    atomicAdd(out + 2, w * (v.z + ev.z));
    atomicAdd(out + 3, w * (v.w + ev.w));
}

// ---------------- host orchestration ----------------

extern "C" void kernel_launch(void* const* d_in, const int* in_sizes, int n_in,
                              void* d_out, int out_size, void* d_ws, size_t ws_size,
                              hipStream_t stream) {
    (void)n_in; (void)out_size; (void)ws_size;

    const float* x = (const float*)d_in[0];
    const int* eidx = (const int*)d_in[1];
    const float* ea = (const float*)d_in[2];
    const float* fW1 = (const float*)d_in[3];
    const float* fW2 = (const float*)d_in[4];
    const float* fb1 = (const float*)d_in[5];
    const float* fb2 = (const float*)d_in[6];

    const int N = in_sizes[0] / HID;   // 50000
    const int E = in_sizes[1] / 2;     // 800000
    const int* srcI = eidx;
    const int* dstI = eidx + E;

    // workspace layout (floats)
    float* ws = (float*)d_ws;
    const size_t fN = (size_t)N * HID;
    float* Q  = ws;
    float* Kf = Q + fN;
    float* Vf = Kf + fN;
    float* Hh = Vf + fN;      // skip + attention accumulation
    float* T1 = Hh + fN;      // MLP hidden
    float* Xb = T1 + fN;      // layer output
    float* Mx = Xb + fN;                   // N*4 segment max
    float* Dn = Mx + (size_t)N * NHEAD;    // N*4 softmax denom
    float* Al = Dn + (size_t)N * NHEAD;    // E*4 alpha / ex
    float* Em = Al + (size_t)E * NHEAD;    // E*128 edge features

    const dim3 blk(256);
    const int gN = (N + 127) / 128;
    const int gE = (E + 127) / 128;
    const int gNH = (N * NHEAD + 255) / 256;
    const int gEH = (E * NHEAD + 255) / 256;
    const long long scatterT = (long long)E * 32;
    const int gSc = (int)((scatterT + 255) / 256);

    const float* xin = x;
    for (int l = 0; l < 2; ++l) {
        // params per layer, jax tree order: W1,W2,We,Wk,Wq,Ws,Wv,b1,b2,bk,bq,bv
        const int p = 7 + l * 12;
        const float* W1 = (const float*)d_in[p + 0];
        const float* W2 = (const float*)d_in[p + 1];
        const float* We = (const float*)d_in[p + 2];
        const float* Wk = (const float*)d_in[p + 3];
        const float* Wq = (const float*)d_in[p + 4];
        const float* Wsk = (const float*)d_in[p + 5];
        const float* Wv = (const float*)d_in[p + 6];
        const float* b1 = (const float*)d_in[p + 7];
        const float* b2 = (const float*)d_in[p + 8];
        const float* bk = (const float*)d_in[p + 9];
        const float* bq = (const float*)d_in[p + 10];
        const float* bv = (const float*)d_in[p + 11];

        gemm_wmma<HID><<<gN, blk, 0, stream>>>(xin, Wq, bq, nullptr, Q, N, 0);
        gemm_wmma<HID><<<gN, blk, 0, stream>>>(xin, Wk, bk, nullptr, Kf, N, 0);
        gemm_wmma<HID><<<gN, blk, 0, stream>>>(xin, Wv, bv, nullptr, Vf, N, 0);
        gemm_wmma<63><<<gE, blk, 0, stream>>>(ea, We, nullptr, nullptr, Em, E, 0);
        // h starts as skip term x @ Ws; attention adds into it
        gemm_wmma<HID><<<gN, blk, 0, stream>>>(xin, Wsk, nullptr, nullptr, Hh, N, 0);

        init_md<<<gNH, blk, 0, stream>>>(Mx, Dn, N * NHEAD);
        attn_alpha<<<gEH, blk, 0, stream>>>(srcI, dstI, Q, Kf, Em, Al, Mx, E);
        attn_exp<<<gEH, blk, 0, stream>>>(dstI, Al, Mx, Dn, E);
        attn_scatter<<<gSc, blk, 0, stream>>>(srcI, dstI, Vf, Em, Al, Dn, Hh, E);

        // residual MLP: x = h + gelu(gelu(h@W1+b1)@W2+b2)
        gemm_wmma<HID><<<gN, blk, 0, stream>>>(Hh, W1, b1, nullptr, T1, N, 1);
        gemm_wmma<HID><<<gN, blk, 0, stream>>>(T1, W2, b2, Hh, Xb, N, 1);
        xin = Xb;
    }

    // final MLP: out = gelu(gelu(x@W1+b1)@W2+b2)
    gemm_wmma<HID><<<gN, blk, 0, stream>>>(Xb, fW1, fb1, nullptr, T1, N, 1);
    gemm_wmma<HID><<<gN, blk, 0, stream>>>(T1, fW2, fb2, nullptr, (float*)d_out, N, 1);
}